// Convolution_63324997812584
// MI455X (gfx1250) — compile-verified
//
#include <hip/hip_runtime.h>

#define N_NODES 50000
#define N_EDGES 800000

typedef float v2f __attribute__((ext_vector_type(2)));
typedef float v8f __attribute__((ext_vector_type(8)));

__device__ __forceinline__ v8f wmma_f32_4(v2f a, v2f b, v8f c) {
  // D = A(16x4) * B(4x16) + C, fp32 WMMA (CDNA5)
  return __builtin_amdgcn_wmma_f32_16x16x4_f32(false, a, false, b, (short)0, c,
                                               false, false);
}

// ---------------------------------------------------------------- zero scratch
__global__ void zero_kernel(float* __restrict__ p, long long n) {
  long long i = (long long)blockIdx.x * blockDim.x + threadIdx.x;
  long long stride = (long long)gridDim.x * blockDim.x;
  for (; i < n; i += stride) p[i] = 0.0f;
}

// ------------------------------------------- pack weights into WMMA frag order
// W1frag[((s*7+t)*32+lane)] = {W1pad[4s+2*(lane>>4)][16t+(lane&15)], next row}
// W2frag[((s*8+t)*32+lane)] = {W2  [4s+2*(lane>>4)][16t+(lane&15)], next row}
__global__ __launch_bounds__(256) void pack_weights_kernel(
    const float* __restrict__ W1, const float* __restrict__ W2,
    float* __restrict__ W1frag, float* __restrict__ W2frag) {
  int i = blockIdx.x * 256 + threadIdx.x;
  if (i < 672) {  // 3 K-steps * 7 N-tiles * 32 lanes
    int lane = i & 31;
    int t = (i >> 5) % 7;
    int s = i / 224;
    int k0 = 4 * s + 2 * (lane >> 4);
    int col = 16 * t + (lane & 15);
    float vx = (k0 < 10 && col < 100) ? W1[k0 * 100 + col] : 0.0f;
    float vy = (k0 + 1 < 10 && col < 100) ? W1[(k0 + 1) * 100 + col] : 0.0f;
    W1frag[2 * i] = vx;
    W1frag[2 * i + 1] = vy;
  }
  if (i < 6400) {  // 25 K-steps * 8 N-tiles * 32 lanes
    int lane = i & 31;
    int t = (i >> 5) & 7;
    int s = i >> 8;
    int k0 = 4 * s + 2 * (lane >> 4);
    int col = 16 * t + (lane & 15);
    W2frag[2 * i] = W2[k0 * 128 + col];
    W2frag[2 * i + 1] = W2[(k0 + 1) * 128 + col];
  }
}

// ------------------------------------------------- node linear: y = lin_l1(x)
__global__ __launch_bounds__(256) void node_l1_kernel(
    const float* __restrict__ x, const float* __restrict__ W0,
    const float* __restrict__ W1, float* __restrict__ y0,
    float* __restrict__ y1) {
  __shared__ float sW0[32 * 32];
  __shared__ float sW1[32 * 32];
  int tid = threadIdx.x;
  for (int i = tid; i < 1024; i += 256) {
    sW0[i] = W0[i];
    sW1[i] = W1[i];
  }
  __syncthreads();

  int g = blockIdx.x * 256 + tid;
  int n = g >> 5;
  int w = g & 31;
  const float* xr = x + (size_t)n * 128;

  float s0 = 0.f, a0 = 0.f, a1 = 0.f, a2 = 0.f;
  for (int u = 0; u < 32; ++u) {
    float w0 = sW0[u * 32 + w];
    float w1 = sW1[u * 32 + w];
    s0 += xr[u] * w0;
    a0 += xr[32 + u * 3 + 0] * w1;
    a1 += xr[32 + u * 3 + 1] * w1;
    a2 += xr[32 + u * 3 + 2] * w1;
  }
  const float isq32 = 0.17677669529663687f;  // 1/sqrt(32)
  y0[(size_t)n * 32 + w] = s0 * isq32;
  float* yp = y1 + (size_t)n * 96 + w * 3;
  yp[0] = a0 * isq32;
  yp[1] = a1 * isq32;
  yp[2] = a2 * isq32;
}

// ----------------------------------------------------------------- edge kernel
// Per wave: 16-edge tile.  w = relu(elen @ Wfc1 / sqrt(10)) @ Wfc2 / sqrt(100)
// then per-edge irrep tensor product, atomic scatter into mid0/mid1.
__global__ __launch_bounds__(128) void edge_kernel(
    const float* __restrict__ elen, const float* __restrict__ W1frag,
    const float* __restrict__ W2frag, const int* __restrict__ esrc,
    const int* __restrict__ edst, const float* __restrict__ attr,
    const float* __restrict__ y0, const float* __restrict__ y1,
    float* __restrict__ mid0, float* __restrict__ mid1) {
  __shared__ __attribute__((aligned(16))) float atile[4][16 * 12];
  __shared__ __attribute__((aligned(16))) float stage[4][16 * 130];

  const int tid = threadIdx.x;
  const int wid = tid >> 5;
  const int lane = tid & 31;
  const int half = lane >> 4;  // 0: K pair {4s,4s+1}, 1: {4s+2,4s+3}
  const int l15 = lane & 15;
  const int e0 = (blockIdx.x * 4 + wid) * 16;

  float* At = &atile[wid][0];
  float* hs = &stage[wid][0];
  const v2f* W1f = (const v2f*)W1frag;
  const v2f* W2f = (const v2f*)W2frag;

  // stage A tile (16 x 10, zero-padded to 16 x 12) into LDS
  const float* Ae = elen + (size_t)e0 * 10;
  #pragma unroll
  for (int i = lane; i < 192; i += 32) {
    int r = i / 12, c = i % 12;
    At[i] = (c < 10) ? Ae[r * 10 + c] : 0.0f;
  }
  __builtin_amdgcn_wave_barrier();

  // ---- GEMM1: 16x12 @ 12x112 (padded), 3 K-steps, 7 N-tiles
  v8f acc[7] = {};
  #pragma unroll
  for (int s = 0; s < 3; ++s) {
    int k0 = 4 * s + 2 * half;
    v2f a = *(const v2f*)&At[l15 * 12 + k0];
    #pragma unroll
    for (int t = 0; t < 7; ++t) {
      v2f b = W1f[(s * 7 + t) * 32 + lane];
      acc[t] = wmma_f32_4(a, b, acc[t]);
    }
  }

  // relu + 1/sqrt(10); C layout: VGPR r -> row (r + 8*half), col 16t+l15
  const float inv_sqrt_nb = 0.31622776601683794f;
  const int rowb = 8 * half;
  #pragma unroll
  for (int t = 0; t < 7; ++t) {
    int colb = 16 * t + l15;
    if (colb < 100) {
      #pragma unroll
      for (int r = 0; r < 8; ++r)
        hs[(rowb + r) * 130 + colb] = fmaxf(acc[t][r], 0.0f) * inv_sqrt_nb;
    }
  }
  __builtin_amdgcn_wave_barrier();

  // ---- GEMM2: 16x100 @ 100x128, K = 25 steps of 4, N = 8 tiles of 16
  v8f d[8] = {};
  for (int s = 0; s < 25; ++s) {
    int k0 = 4 * s + 2 * half;
    v2f a = *(const v2f*)&hs[l15 * 130 + k0];
    const v2f* bp = W2f + (size_t)(s * 8) * 32 + lane;
    #pragma unroll
    for (int t = 0; t < 8; ++t) {
      v2f b = bp[t * 32];
      d[t] = wmma_f32_4(a, b, d[t]);
    }
  }
  __builtin_amdgcn_wave_barrier();

  // stage w = d / sqrt(100)
  #pragma unroll
  for (int t = 0; t < 8; ++t) {
    int colb = 16 * t + l15;
    #pragma unroll
    for (int r = 0; r < 8; ++r)
      hs[(rowb + r) * 130 + colb] = d[t][r] * 0.1f;
  }
  __builtin_amdgcn_wave_barrier();

  // ---- per-edge irrep tensor product + atomic scatter. lane = channel u.
  const float INV_SQRT3 = 0.5773502691896258f;
  const int u = lane;
  for (int e = 0; e < 16; ++e) {
    int eg = e0 + e;
    int src = esrc[eg];
    int dst = edst[eg];
    const float* ap = attr + (size_t)eg * 4;
    float sh0 = ap[0], s1x = ap[1], s1y = ap[2], s1z = ap[3];

    float wA = hs[e * 130 + u];
    float wB = hs[e * 130 + 32 + u];
    float wC = hs[e * 130 + 64 + u];
    float wD = hs[e * 130 + 96 + u];

    float g0 = y0[(size_t)src * 32 + u];
    const float* g1p = y1 + (size_t)src * 96 + u * 3;
    float g1x = g1p[0], g1y = g1p[1], g1z = g1p[2];
    float dotg = g1x * s1x + g1y * s1y + g1z * s1z;

    float* m0 = mid0 + (size_t)dst * 64;
    float* m1 = mid1 + (size_t)dst * 192;
    unsafeAtomicAdd(&m0[u], wA * g0 * sh0);
    unsafeAtomicAdd(&m0[32 + u], wD * dotg * INV_SQRT3);
    unsafeAtomicAdd(&m1[u * 3 + 0], wB * g0 * s1x);
    unsafeAtomicAdd(&m1[u * 3 + 1], wB * g0 * s1y);
    unsafeAtomicAdd(&m1[u * 3 + 2], wB * g0 * s1z);
    unsafeAtomicAdd(&m1[96 + u * 3 + 0], wC * g1x * sh0);
    unsafeAtomicAdd(&m1[96 + u * 3 + 1], wC * g1y * sh0);
    unsafeAtomicAdd(&m1[96 + u * 3 + 2], wC * g1z * sh0);
  }
}

// ---------------------------------------------------- final: si + 0.5*lin_l2
__global__ __launch_bounds__(256) void final_kernel(
    const float* __restrict__ x, const float* __restrict__ Wsi0,
    const float* __restrict__ Wsi1, const float* __restrict__ Wl20,
    const float* __restrict__ Wl21, const float* __restrict__ mid0,
    const float* __restrict__ mid1, float* __restrict__ out) {
  __shared__ float s0[32 * 32], s1[32 * 32];
  __shared__ float s2[64 * 32], s3[64 * 32];
  int tid = threadIdx.x;
  for (int i = tid; i < 1024; i += 256) {
    s0[i] = Wsi0[i];
    s1[i] = Wsi1[i];
  }
  for (int i = tid; i < 2048; i += 256) {
    s2[i] = Wl20[i];
    s3[i] = Wl21[i];
  }
  __syncthreads();

  int g = blockIdx.x * 256 + tid;
  int n = g >> 5;
  int w = g & 31;
  const float* xr = x + (size_t)n * 128;

  float si0 = 0.f, a0 = 0.f, a1 = 0.f, a2 = 0.f;
  for (int u = 0; u < 32; ++u) {
    float w0 = s0[u * 32 + w];
    float w1 = s1[u * 32 + w];
    si0 += xr[u] * w0;
    a0 += xr[32 + u * 3 + 0] * w1;
    a1 += xr[32 + u * 3 + 1] * w1;
    a2 += xr[32 + u * 3 + 2] * w1;
  }
  const float isq32 = 0.17677669529663687f;
  si0 *= isq32;
  a0 *= isq32;
  a1 *= isq32;
  a2 *= isq32;

  const float* m0 = mid0 + (size_t)n * 64;
  const float* m1 = mid1 + (size_t)n * 192;
  float o0 = 0.f, b0 = 0.f, b1 = 0.f, b2 = 0.f;
  for (int u = 0; u < 64; ++u) {
    float w0 = s2[u * 32 + w];
    float w1 = s3[u * 32 + w];
    o0 += m0[u] * w0;
    b0 += m1[u * 3 + 0] * w1;
    b1 += m1[u * 3 + 1] * w1;
    b2 += m1[u * 3 + 2] * w1;
  }
  // 0.5 * inv_sqrt_nn(1/4) * 1/sqrt(64)(1/8) = 1/64
  const float k = 0.015625f;
  float* op = out + (size_t)n * 128;
  op[w] = si0 + k * o0;
  op[32 + w * 3 + 0] = a0 + k * b0;
  op[32 + w * 3 + 1] = a1 + k * b1;
  op[32 + w * 3 + 2] = a2 + k * b2;
}

extern "C" void kernel_launch(void* const* d_in, const int* in_sizes, int n_in,
                              void* d_out, int out_size, void* d_ws,
                              size_t ws_size, hipStream_t stream) {
  const float* x = (const float*)d_in[0];
  const int* esrc = (const int*)d_in[1];
  const int* edst = (const int*)d_in[2];
  const float* attr = (const float*)d_in[3];
  const float* elen = (const float*)d_in[4];
  const float* Wsi0 = (const float*)d_in[5];
  const float* Wsi1 = (const float*)d_in[6];
  const float* Wl10 = (const float*)d_in[7];
  const float* Wl11 = (const float*)d_in[8];
  const float* Wfc1 = (const float*)d_in[9];
  const float* Wfc2 = (const float*)d_in[10];
  const float* Wl20 = (const float*)d_in[11];
  const float* Wl21 = (const float*)d_in[12];
  float* out = (float*)d_out;

  float* ws = (float*)d_ws;
  float* y0 = ws;                             // N*32
  float* y1 = ws + (size_t)N_NODES * 32;      // N*96
  float* mid0 = ws + (size_t)N_NODES * 128;   // N*64
  float* mid1 = ws + (size_t)N_NODES * 192;   // N*192
  float* w1frag = ws + (size_t)N_NODES * 384; // 1344 floats
  float* w2frag = w1frag + 1344;              // 12800 floats

  zero_kernel<<<2048, 256, 0, stream>>>(mid0, (long long)N_NODES * 256);
  pack_weights_kernel<<<25, 256, 0, stream>>>(Wfc1, Wfc2, w1frag, w2frag);
  node_l1_kernel<<<(N_NODES * 32) / 256, 256, 0, stream>>>(x, Wl10, Wl11, y0,
                                                           y1);
  edge_kernel<<<N_EDGES / 64, 128, 0, stream>>>(elen, w1frag, w2frag, esrc,
                                                edst, attr, y0, y1, mid0, mid1);
  final_kernel<<<(N_NODES * 32) / 256, 256, 0, stream>>>(
      x, Wsi0, Wsi1, Wl20, Wl21, mid0, mid1, out);
}